// ImprovedTripletLoss_81544249082160
// MI455X (gfx1250) — compile-verified
//
#include <hip/hip_runtime.h>
#include <hip/hip_bf16.h>

typedef __attribute__((ext_vector_type(2))) float v2f;
typedef __attribute__((ext_vector_type(4))) float v4f;
typedef __attribute__((ext_vector_type(8))) float v8f;

#define N_EMB 8192
#define D_EMB 512
#define TILE  128
#define KT    32
#define LDSS  36            // padded LDS row stride (floats): conflict-free, 16B-aligned
#define NPANEL (D_EMB / KT) // 16

// ---------------- Kernel 1: L2 normalize rows, emit squared norm ----------------
__global__ __launch_bounds__(256)
void tl_normalize(const float* __restrict__ emb,
                  float* __restrict__ embN,
                  float* __restrict__ sq) {
    const int wave = threadIdx.x >> 5;
    const int lane = threadIdx.x & 31;
    const int row  = blockIdx.x * 8 + wave;
    const float* src = emb + (size_t)row * D_EMB;
    const int base = lane * 16;

    v4f vals[4];
    float s = 0.0f;
#pragma unroll
    for (int i = 0; i < 4; ++i) {
        vals[i] = *(const v4f*)&src[base + 4 * i];
#pragma unroll
        for (int e = 0; e < 4; ++e) s += vals[i][e] * vals[i][e];
    }
#pragma unroll
    for (int m = 1; m < 32; m <<= 1) s += __shfl_xor(s, m, 32);

    const float nrm   = sqrtf(s);
    const float scale = 1.0f / fmaxf(nrm, 1e-12f);

    float* dst = embN + (size_t)row * D_EMB;
#pragma unroll
    for (int i = 0; i < 4; ++i) {
        v4f o;
#pragma unroll
        for (int e = 0; e < 4; ++e) o[e] = vals[i][e] * scale;
        *(v4f*)&dst[base + 4 * i] = o;
    }
    if (lane == 0) sq[row] = s * scale * scale;  // == diag of Gram (|e_n|^2)
}

// ---------------- Kernel 2: init per-anchor accumulators ----------------
__global__ __launch_bounds__(256)
void tl_init(unsigned int* __restrict__ hpBits, unsigned int* __restrict__ hnBits) {
    const int i = blockIdx.x * 256 + threadIdx.x;
    if (i < N_EMB) {
        hpBits[i] = 0u;                        // 0.0f  (dist is always > 0)
        hnBits[i] = __float_as_uint(1e9f);     // reference "big"
    }
}

// Issue global loads of one 128xKT panel pair into registers (no wait here).
__device__ __forceinline__ void tl_load_panel(const float* __restrict__ embN,
                                              int rowBase, int colBase, int k0,
                                              int tid, v4f ra[4], v4f rb[4]) {
#pragma unroll
    for (int i = 0; i < 4; ++i) {
        const int idx = tid + 256 * i;        // 0..1023 over (128 rows x 8 float4)
        const int r   = idx >> 3;
        const int c4  = (idx & 7) << 2;
        ra[i] = *(const v4f*)&embN[(size_t)(rowBase + r) * D_EMB + k0 + c4];
        rb[i] = *(const v4f*)&embN[(size_t)(colBase + r) * D_EMB + k0 + c4];
    }
}

// ---------------- Kernel 3: fused WMMA Gram (upper triangle) + hardest pos/neg ----------------
__global__ __launch_bounds__(256)
void tl_gemm_reduce(const float* __restrict__ embN,
                    const float* __restrict__ sq,
                    const int*   __restrict__ labels,
                    unsigned int* __restrict__ hpBits,
                    unsigned int* __restrict__ hnBits) {
    if (blockIdx.y > blockIdx.x) return;       // symmetry: upper-triangular blocks only
    const bool offdiag = (blockIdx.y != blockIdx.x);

    __shared__ float As[TILE * LDSS];
    __shared__ float Bs[TILE * LDSS];
    __shared__ unsigned int hpT[TILE];         // transposed-direction per-column partials
    __shared__ unsigned int hnT[TILE];

    const int tid  = threadIdx.x;
    const int wave = tid >> 5;
    const int lane = tid & 31;
    const int half = lane >> 4;   // 0: rows v, 1: rows v+8 of the 16x16 tile
    const int l16  = lane & 15;   // column within tile

    const int rowBase = blockIdx.y * TILE;
    const int colBase = blockIdx.x * TILE;

    if (tid < TILE) {
        hpT[tid] = 0u;
        hnT[tid] = __float_as_uint(1e9f);
    }

    v8f acc[8];
#pragma unroll
    for (int t = 0; t < 8; ++t)
#pragma unroll
        for (int e = 0; e < 8; ++e) acc[t][e] = 0.0f;

    const int rA = 16 * wave + l16;  // this wave's A-fragment row in the LDS panel

    v4f ra[4], rb[4];
    tl_load_panel(embN, rowBase, colBase, 0, tid, ra, rb);

    for (int p = 0; p < NPANEL; ++p) {
        // Commit staged registers (waits on their loads) into LDS.
#pragma unroll
        for (int i = 0; i < 4; ++i) {
            const int idx = tid + 256 * i;
            const int r   = idx >> 3;
            const int c4  = (idx & 7) << 2;
            *(v4f*)&As[r * LDSS + c4] = ra[i];
            *(v4f*)&Bs[r * LDSS + c4] = rb[i];
        }
        __syncthreads();

        // Issue next panel's global loads; latency hides behind the WMMA block.
        if (p + 1 < NPANEL)
            tl_load_panel(embN, rowBase, colBase, (p + 1) * KT, tid, ra, rb);

#pragma unroll
        for (int kk = 0; kk < KT; kk += 4) {
            const int kcol = kk + 2 * half;
            v2f a = *(const v2f*)&As[rA * LDSS + kcol];
#pragma unroll
            for (int ct = 0; ct < 8; ++ct) {
                v2f b = *(const v2f*)&Bs[(16 * ct + l16) * LDSS + kcol];
                acc[ct] = __builtin_amdgcn_wmma_f32_16x16x4_f32(
                    false, a, false, b, (short)0, acc[ct], false, false);
            }
        }
        __syncthreads();
    }

    // ---- Epilogue: distances + masked hardest pos/neg, both anchor directions ----
    const int rStripe = rowBase + 16 * wave;
    int   labR[8];
    float sqR[8];
#pragma unroll
    for (int v = 0; v < 8; ++v) {
        const int r = rStripe + v + 8 * half;
        labR[v] = labels[r];
        sqR[v]  = sq[r];
    }

    float hp[8], hn[8];
#pragma unroll
    for (int v = 0; v < 8; ++v) { hp[v] = -1.0f; hn[v] = 1e30f; }

#pragma unroll
    for (int ct = 0; ct < 8; ++ct) {
        const int   c    = colBase + 16 * ct + l16;
        const int   labC = labels[c];
        const float sqC  = sq[c];
        float pT = -1.0f, nT = 1e30f;          // anchor = column c (transposed dir)
#pragma unroll
        for (int v = 0; v < 8; ++v) {
            const int   r    = rStripe + v + 8 * half;
            const float g    = acc[ct][v];
            const float d2   = sqC - 2.0f * g + sqR[v];
            const float dist = sqrtf(fmaxf(d2, 0.0f) + 1e-12f);
            const bool  same = (labR[v] == labC);
            const bool  pok  = same && (r != c);
            const float pv = pok  ? dist : -1.0f;
            const float nv = same ? 1e30f : dist;
            hp[v] = fmaxf(hp[v], pv);          // anchor = row r
            hn[v] = fminf(hn[v], nv);
            pT = fmaxf(pT, pv);
            nT = fminf(nT, nv);
        }
        if (offdiag) {
            // combine the two half-waves (rows v and v+8), then merge waves via LDS
            pT = fmaxf(pT, __shfl_xor(pT, 16, 32));
            nT = fminf(nT, __shfl_xor(nT, 16, 32));
            if (half == 0) {
                if (pT > 0.0f)  atomicMax(&hpT[16 * ct + l16], __float_as_uint(pT));
                if (nT < 1e29f) atomicMin(&hnT[16 * ct + l16], __float_as_uint(nT));
            }
        }
    }

    // normal direction: reduce over the 16 columns held by this half-wave
#pragma unroll
    for (int v = 0; v < 8; ++v) {
#pragma unroll
        for (int m = 1; m < 16; m <<= 1) {
            hp[v] = fmaxf(hp[v], __shfl_xor(hp[v], m, 32));
            hn[v] = fminf(hn[v], __shfl_xor(hn[v], m, 32));
        }
    }
    if (l16 == 0) {
#pragma unroll
        for (int v = 0; v < 8; ++v) {
            const int r = rStripe + v + 8 * half;
            if (hp[v] > 0.0f)  // nonneg float bits order == uint order
                atomicMax(&hpBits[r], __float_as_uint(hp[v]));
            if (hn[v] < 1e29f)
                atomicMin(&hnBits[r], __float_as_uint(hn[v]));
        }
    }

    if (offdiag) {
        __syncthreads();
        if (tid < TILE) {
            if (hpT[tid] != 0u)
                atomicMax(&hpBits[colBase + tid], hpT[tid]);
            if (hnT[tid] < __float_as_uint(1e9f))
                atomicMin(&hnBits[colBase + tid], hnT[tid]);
        }
    }
}

// ---------------- Kernel 4: final loss ----------------
__global__ __launch_bounds__(256)
void tl_finalize(const unsigned int* __restrict__ hpBits,
                 const unsigned int* __restrict__ hnBits,
                 float* __restrict__ out) {
    __shared__ float ssum[256];
    __shared__ int   scnt[256];
    float sum = 0.0f;
    int   cnt = 0;
    for (int i = threadIdx.x; i < N_EMB; i += 256) {
        const float hp = __uint_as_float(hpBits[i]);
        const float hn = __uint_as_float(hnBits[i]);
        const bool valid = (hp > 0.0f) && (hn < 1e8f);  // dist <= 2, so init 1e9 means "none"
        if (valid) {
            sum += fmaxf(hp - hn + 1.0f, 0.0f);
            cnt += 1;
        }
    }
    ssum[threadIdx.x] = sum;
    scnt[threadIdx.x] = cnt;
    __syncthreads();
    for (int s = 128; s > 0; s >>= 1) {
        if (threadIdx.x < s) {
            ssum[threadIdx.x] += ssum[threadIdx.x + s];
            scnt[threadIdx.x] += scnt[threadIdx.x + s];
        }
        __syncthreads();
    }
    if (threadIdx.x == 0) {
        const int nv = scnt[0] > 1 ? scnt[0] : 1;
        out[0] = ssum[0] / (float)nv;
    }
}

extern "C" void kernel_launch(void* const* d_in, const int* in_sizes, int n_in,
                              void* d_out, int out_size, void* d_ws, size_t ws_size,
                              hipStream_t stream) {
    const float* emb    = (const float*)d_in[0];
    const int*   labels = (const int*)d_in[1];
    float*       out    = (float*)d_out;

    char* ws = (char*)d_ws;
    float*        embN   = (float*)ws;                                   // 16 MB
    float*        sq     = (float*)(ws + (size_t)N_EMB * D_EMB * 4);     // 32 KB
    unsigned int* hpBits = (unsigned int*)(ws + (size_t)N_EMB * D_EMB * 4 + N_EMB * 4);
    unsigned int* hnBits = hpBits + N_EMB;

    tl_normalize<<<N_EMB / 8, 256, 0, stream>>>(emb, embN, sq);
    tl_init<<<N_EMB / 256, 256, 0, stream>>>(hpBits, hnBits);
    dim3 grid(N_EMB / TILE, N_EMB / TILE);   // lower-triangle blocks exit immediately
    tl_gemm_reduce<<<grid, 256, 0, stream>>>(embN, sq, labels, hpBits, hnBits);
    tl_finalize<<<1, 256, 0, stream>>>(hpBits, hnBits, out);
}